// GraphSAGEWithEmbeddings_35296041239118
// MI455X (gfx1250) — compile-verified
//
#include <hip/hip_runtime.h>

// ---------------------------------------------------------------------------
// GraphSAGE (mean) 2-layer + classifier for MI455X (gfx1250, wave32).
//
// deg[d]   = sum_e 1[dst==d]
// sum1[d]  = sum_e x[src]                (11 wide)
// h        = relu(sum1/deg @ W1_l + x @ W1_r + b1)        [N,64]
// p        = h @ W2_l   (WMMA f32 16x16x4)                [N,32]
// sum2[d]  = sum_e p[src]                (32 wide)
// q        = h @ W2_r   (WMMA f32 16x16x4)                [N,32]
// emb      = relu(sum2/deg + q + b2)                      [N,32]
// logits   = emb @ Wc + bc                                [N,3]
// d_out    = [logits | emb]
// ---------------------------------------------------------------------------

#define NNODES 100000

typedef float v2f __attribute__((ext_vector_type(2)));
typedef float v8f __attribute__((ext_vector_type(8)));

// ---------------- zero scratch (deg | sum1 | sum2 are contiguous) ----------
__global__ void sage_zero_ws(float* __restrict__ w, size_t count) {
  size_t i = (size_t)blockIdx.x * blockDim.x + threadIdx.x;
  size_t stride = (size_t)gridDim.x * blockDim.x;
  for (; i < count; i += stride) w[i] = 0.0f;
}

// ---------------- edge pass 1: scatter x (11 wide) + degree ----------------
__global__ void sage_edge1(const int* __restrict__ src, const int* __restrict__ dst,
                           const float* __restrict__ x, float* __restrict__ sum1,
                           float* __restrict__ deg, int E) {
  int e = blockIdx.x * blockDim.x + threadIdx.x;
  if (e >= E) return;
  int s = src[e];
  int d = dst[e];
  atomicAdd(&deg[d], 1.0f);
  const float* xs = x + (size_t)s * 11;
  float* sd = sum1 + (size_t)d * 11;
#pragma unroll
  for (int f = 0; f < 11; ++f) atomicAdd(&sd[f], xs[f]);
}

// ---------------- node transform 1: h = relu([agg|x] @ [W1l;W1r] + b1) -----
__global__ void sage_node1(const float* __restrict__ x, const float* __restrict__ sum1,
                           const float* __restrict__ deg,
                           const float* __restrict__ W1l, const float* __restrict__ W1r,
                           const float* __restrict__ b1, float* __restrict__ h) {
  __shared__ float agg[11];
  __shared__ float xr[11];
  const int node = blockIdx.x;          // one 64-thread block per node
  const int t = threadIdx.x;            // 0..63 = output channel
  if (t < 11) {
    float dv = fmaxf(deg[node], 1.0f);
    agg[t] = sum1[(size_t)node * 11 + t] / dv;
    xr[t]  = x[(size_t)node * 11 + t];
  }
  __syncthreads();
  float acc = b1[t];
#pragma unroll
  for (int k = 0; k < 11; ++k) {
    acc = fmaf(agg[k], W1l[k * 64 + t], acc);   // W1l: [11,64] row-major
    acc = fmaf(xr[k],  W1r[k * 64 + t], acc);
  }
  h[(size_t)node * 64 + t] = fmaxf(acc, 0.0f);
}

// ---------------- WMMA GEMM: Y[N,32] = X[N,64] @ W[64,32], f32 -------------
// One wave (32 threads) per 16-row tile. N % 16 == 0 guaranteed (100000).
// A layout (16x4 f32, 2 VGPRs/lane): lane<16  -> row=lane,    K = k0, k0+1
//                                    lane>=16 -> row=lane-16, K = k0+2, k0+3
// B layout mirrors A with N=lane. C/D: VGPR v holds M=v (lanes 0-15) and
// M=v+8 (lanes 16-31), N = lane&15.
__global__ void sage_gemm64x32_wmma(const float* __restrict__ X,
                                    const float* __restrict__ W,
                                    float* __restrict__ Y) {
  const int lane  = threadIdx.x & 31;
  const int rbase = blockIdx.x * 16;
  const int m     = lane & 15;
  const int kSel  = (lane >> 4) << 1;       // 0 or 2
  const int row   = rbase + m;

  v8f acc0 = {};                            // output cols 0..15
  v8f acc1 = {};                            // output cols 16..31
  const float* xrow = X + (size_t)row * 64;

#pragma unroll
  for (int k0 = 0; k0 < 64; k0 += 4) {
    v2f a;
    a.x = xrow[k0 + kSel];
    a.y = xrow[k0 + kSel + 1];
    const float* w0 = W + (size_t)(k0 + kSel) * 32;   // W row-major [64,32]
    const float* w1 = w0 + 32;
    v2f b0, b1;
    b0.x = w0[m];       b0.y = w1[m];
    b1.x = w0[m + 16];  b1.y = w1[m + 16];
    acc0 = __builtin_amdgcn_wmma_f32_16x16x4_f32(false, a, false, b0,
                                                 (short)0, acc0, false, false);
    acc1 = __builtin_amdgcn_wmma_f32_16x16x4_f32(false, a, false, b1,
                                                 (short)0, acc1, false, false);
  }

  const int mr = (lane >> 4) << 3;          // 0 or 8
#pragma unroll
  for (int v = 0; v < 8; ++v) {
    const size_t r = (size_t)(rbase + mr + v) * 32;
    Y[r + m]      = acc0[v];
    Y[r + m + 16] = acc1[v];
  }
}

// ---------------- edge pass 2: scatter p (32 wide), 8 threads/edge ---------
__global__ void sage_edge2(const int* __restrict__ src, const int* __restrict__ dst,
                           const float* __restrict__ p, float* __restrict__ sum2, int E) {
  int t = blockIdx.x * blockDim.x + threadIdx.x;
  int e = t >> 3;
  if (e >= E) return;
  int c = (t & 7) << 2;
  int s = src[e];
  int d = dst[e];
  const float4 v = *(const float4*)(p + (size_t)s * 32 + c);
  float* o = sum2 + (size_t)d * 32 + c;
  atomicAdd(o + 0, v.x);
  atomicAdd(o + 1, v.y);
  atomicAdd(o + 2, v.z);
  atomicAdd(o + 3, v.w);
}

// ---------------- final: emb = relu(sum2/deg + q + b2); logits = emb@Wc+bc -
__global__ void sage_node2(const float* __restrict__ sum2, const float* __restrict__ q,
                           const float* __restrict__ deg, const float* __restrict__ b2,
                           const float* __restrict__ Wc, const float* __restrict__ bc,
                           float* __restrict__ out_logits, float* __restrict__ out_emb,
                           int n) {
  int node = blockIdx.x * blockDim.x + threadIdx.x;
  if (node >= n) return;
  float inv = 1.0f / fmaxf(deg[node], 1.0f);
  float l0 = bc[0], l1 = bc[1], l2 = bc[2];
  const float* s2 = sum2 + (size_t)node * 32;
  const float* qr = q + (size_t)node * 32;
  float* er = out_emb + (size_t)node * 32;
#pragma unroll
  for (int c = 0; c < 32; ++c) {
    float e = fmaxf(fmaf(s2[c], inv, qr[c] + b2[c]), 0.0f);
    er[c] = e;
    l0 = fmaf(e, Wc[c * 3 + 0], l0);   // Wc: [32,3] row-major
    l1 = fmaf(e, Wc[c * 3 + 1], l1);
    l2 = fmaf(e, Wc[c * 3 + 2], l2);
  }
  float* lr = out_logits + (size_t)node * 3;
  lr[0] = l0; lr[1] = l1; lr[2] = l2;
}

extern "C" void kernel_launch(void* const* d_in, const int* in_sizes, int n_in,
                              void* d_out, int out_size, void* d_ws, size_t ws_size,
                              hipStream_t stream) {
  const float* x   = (const float*)d_in[0];
  const int*   ei  = (const int*)d_in[1];   // edge_index [2,E] (int32 on device)
  const float* W1l = (const float*)d_in[2];
  const float* W1r = (const float*)d_in[3];
  const float* b1  = (const float*)d_in[4];
  const float* W2l = (const float*)d_in[5];
  const float* W2r = (const float*)d_in[6];
  const float* b2  = (const float*)d_in[7];
  const float* Wc  = (const float*)d_in[8];
  const float* bc  = (const float*)d_in[9];

  const int N = NNODES;
  const int E = in_sizes[1] / 2;
  const int* src = ei;
  const int* dst = ei + E;

  // workspace layout (floats): deg[N] | sum1[11N] | sum2[32N] | h[64N] | p[32N] | q[32N]
  float* ws   = (float*)d_ws;
  float* deg  = ws;
  float* sum1 = ws + (size_t)N;
  float* sum2 = ws + (size_t)12 * N;
  float* h    = ws + (size_t)44 * N;
  float* p    = ws + (size_t)108 * N;
  float* q    = ws + (size_t)140 * N;

  float* out_logits = (float*)d_out;
  float* out_emb    = (float*)d_out + (size_t)N * 3;

  // 1) zero the accumulation regions (deg+sum1+sum2 contiguous = 44N floats)
  sage_zero_ws<<<2048, 256, 0, stream>>>(ws, (size_t)44 * N);

  // 2) layer-1 edge scatter: sum1 += x[src], deg += 1
  sage_edge1<<<(E + 255) / 256, 256, 0, stream>>>(src, dst, x, sum1, deg, E);

  // 3) h = relu(mean-agg @ W1_l + x @ W1_r + b1)
  sage_node1<<<N, 64, 0, stream>>>(x, sum1, deg, W1l, W1r, b1, h);

  // 4) p = h @ W2_l (pre-multiply so we aggregate 32-wide instead of 64-wide)
  sage_gemm64x32_wmma<<<N / 16, 32, 0, stream>>>(h, W2l, p);

  // 5) layer-2 edge scatter: sum2 += p[src]
  sage_edge2<<<((E * 8) + 255) / 256, 256, 0, stream>>>(src, dst, p, sum2, E);

  // 6) q = h @ W2_r
  sage_gemm64x32_wmma<<<N / 16, 32, 0, stream>>>(h, W2r, q);

  // 7) emb = relu(sum2/deg + q + b2); logits = emb @ Wc + bc
  sage_node2<<<(N + 255) / 256, 256, 0, stream>>>(sum2, q, deg, b2, Wc, bc,
                                                  out_logits, out_emb, N);
}